// AffineTransform_79783312490707
// MI455X (gfx1250) — compile-verified
//
#include <hip/hip_runtime.h>
#include <hip/hip_bf16.h>

// Affine bilinear warp, fp32, (B=8, C=3, H=1024, W=1024).
// Roofline: ~201 MB min traffic @ 23.3 TB/s -> ~8.6 us floor; ~0.012 FLOP/byte
// -> no matmul structure (WMMA inapplicable).
// This version cuts per-pixel VALU: interior pixels take
// a clamp-free fast path where all 12 gathers share ONE per-lane
// 64-bit base address and use 24-bit immediate offsets (c*4MB + {0,4,4096,4100}).

namespace {
constexpr int Bv = 8;
constexpr int Cv = 3;
constexpr int Hv = 1024;
constexpr int Wv = 1024;
constexpr int HWv = Hv * Wv;
constexpr int BLOCK = 256;  // 8 wave32 waves
}

__global__ __launch_bounds__(BLOCK)
void affine_bilinear_f32(const float* __restrict__ img,
                         const float* __restrict__ affine,
                         float* __restrict__ out) {
  const int w = blockIdx.x * BLOCK + threadIdx.x;  // lanes contiguous in W
  const int h = blockIdx.y;                        // uniform per block
  const int b = blockIdx.z;                        // uniform per block

  // Uniform per-batch 3x3 affine -> SGPR s_loads.
  const float* A = affine + b * 9;
  const float a00 = A[0], a01 = A[1], a02 = A[2];
  const float a10 = A[3], a11 = A[4], a12 = A[5];

  constexpr float cH = (float)(Hv - 1) * 0.5f;  // 511.5
  constexpr float cW = (float)(Wv - 1) * 0.5f;  // 511.5

  // gx is block-uniform -> the a*gx terms scalarize onto the SALU float pipe.
  const float gx = (float)h - cH;
  const float gy = (float)w - cW;
  const float sx = fmaf(a01, gy, fmaf(a00, gx, a02) + cH);
  const float sy = fmaf(a11, gy, fmaf(a10, gx, a12) + cW);

  const float x0f = floorf(sx);
  const float y0f = floorf(sy);
  const float wx1 = sx - x0f;
  const float wy1 = sy - y0f;
  const float wx0 = 1.0f - wx1;
  const float wy0 = 1.0f - wy1;

  const int x0 = (int)x0f;
  const int y0 = (int)y0f;

  // Output base anchored at channel 1; stores at imm offsets c*HWv, c in {-1,0,1}.
  float* dbase = out + ((size_t)b * Cv + 1) * HWv + (size_t)h * Wv + (size_t)w;

  const bool interior =
      ((unsigned)x0 <= (unsigned)(Hv - 2)) && ((unsigned)y0 <= (unsigned)(Wv - 2));

  if (interior) {
    // ---- Fast path: all 4 corners in bounds; no clamps, no masks. ----
    const float w00 = wx0 * wy0;
    const float w01 = wx0 * wy1;
    const float w10 = wx1 * wy0;
    const float w11 = wx1 * wy1;

    // One per-lane base (channel 1, corner 00). All 12 gathers are imm-offset:
    // bytes = c*4194304 + {0, 4, 4096, 4100}, within signed 24-bit IOFFSET.
    const float* base = img + ((size_t)b * Cv + 1) * HWv + (x0 * Wv + y0);

    // Warm L2 for the next row pair (gfx1250 global_prefetch_b8, speculative).
    __builtin_prefetch(base + 2 * Wv, 0, 0);

#pragma unroll
    for (int c = -1; c <= 1; ++c) {
      const float* p = base + c * HWv;
      const float g00 = p[0];
      const float g01 = p[1];        // adjacent -> mergeable into b64 load
      const float g10 = p[Wv];
      const float g11 = p[Wv + 1];   // adjacent -> mergeable into b64 load
      const float v = fmaf(w00, g00, fmaf(w01, g01, fmaf(w10, g10, w11 * g11)));
      __builtin_nontemporal_store(v, dbase + c * HWv);  // write-once -> NT
    }
  } else {
    // ---- Slow path (image border band): zero-padding via masked weights. ----
    const int x1 = x0 + 1;
    const int y1 = y0 + 1;

    const bool bx0 = (x0 >= 0) && (x0 <= Hv - 1);
    const bool bx1 = (x1 >= 0) && (x1 <= Hv - 1);
    const bool by0 = (y0 >= 0) && (y0 <= Wv - 1);
    const bool by1 = (y1 >= 0) && (y1 <= Wv - 1);

    const float m00 = (bx0 && by0) ? wx0 * wy0 : 0.0f;
    const float m01 = (bx0 && by1) ? wx0 * wy1 : 0.0f;
    const float m10 = (bx1 && by0) ? wx1 * wy0 : 0.0f;
    const float m11 = (bx1 && by1) ? wx1 * wy1 : 0.0f;

    const int xc0 = x0 < 0 ? 0 : (x0 > Hv - 1 ? Hv - 1 : x0);
    const int xc1 = x1 < 0 ? 0 : (x1 > Hv - 1 ? Hv - 1 : x1);
    const int yc0 = y0 < 0 ? 0 : (y0 > Wv - 1 ? Wv - 1 : y0);
    const int yc1 = y1 < 0 ? 0 : (y1 > Wv - 1 ? Wv - 1 : y1);

    const int i00 = xc0 * Wv + yc0;
    const int i01 = xc0 * Wv + yc1;
    const int i10 = xc1 * Wv + yc0;
    const int i11 = xc1 * Wv + yc1;

    const float* src = img + (size_t)b * Cv * HWv;
#pragma unroll
    for (int c = 0; c < Cv; ++c) {
      const float* p = src + (size_t)c * HWv;
      const float v =
          fmaf(m00, p[i00], fmaf(m01, p[i01], fmaf(m10, p[i10], m11 * p[i11])));
      __builtin_nontemporal_store(v, dbase + (c - 1) * HWv);
    }
  }
}

extern "C" void kernel_launch(void* const* d_in, const int* in_sizes, int n_in,
                              void* d_out, int out_size, void* d_ws, size_t ws_size,
                              hipStream_t stream) {
  const float* img = (const float*)d_in[0];     // (8, 3, 1024, 1024) fp32
  const float* affine = (const float*)d_in[1];  // (8, 3, 3) fp32
  float* out = (float*)d_out;                   // (8, 3, 1024, 1024) fp32

  dim3 grid(Wv / BLOCK, Hv, Bv);  // (4, 1024, 8)
  dim3 block(BLOCK, 1, 1);
  affine_bilinear_f32<<<grid, block, 0, stream>>>(img, affine, out);
}